// PINN_MOE_53309134078333
// MI455X (gfx1250) — compile-verified
//
#include <hip/hip_runtime.h>

typedef __attribute__((ext_vector_type(2))) float v2f;
typedef __attribute__((ext_vector_type(8))) float v8f;

#define WAVES    4
#define TPB      (WAVES * 32)
#define ROWS_PB  (WAVES * 16)

// ---- LDS layout (float offsets) ----
// Weights, zero-padded to [Kpad][16] (or [Kpad][32] for Wp1) so pad cols/rows give 0.
#define WC   0        // 12x16  fused attention 10->11
#define BCV  192      // 16
#define EW1  208      // 3 * 12*16
#define EB1  784      // 3 * 16
#define EW2  832      // 3 * 4*16
#define EB2  1024
#define EW3  1072     // 3 * 8*16
#define EB3  1456
#define EW4  1504     // 3 * 16*16
#define EB4  2272
#define EW5  2320     // 3 * 8*16
#define EB5  2704
#define WP1  2752     // 8*32  (K pad 6->8)
#define BP1  3008     // 32
#define WP2  3040     // 32*16
#define BP2  3552     // 16
#define WP3  3568     // 16*16 (only col 0 nonzero)
#define BP3  3824     // 16
#define WTOT 3840
#define WAVE_SCR 1200 // per-wave: TA 272 | TB 272 | TW 528 | SC 128

// Wave-local LDS fence: all tiles are wave-private, and same-wave LDS ops
// complete in order (ISA 7.3), so a cross-lane RAW through LDS needs only
// s_wait_dscnt 0 plus a compiler scheduling fence -- no block barrier.
__device__ __forceinline__ void wave_sync() {
  __builtin_amdgcn_wave_barrier();
  asm volatile("s_wait_dscnt 0" ::: "memory");
  __builtin_amdgcn_wave_barrier();
}

// One 16x16 (M=samples, N=out-features) tile-GEMM accumulated over kChunks of K=4,
// using V_WMMA_F32_16X16X4_F32. A read from LDS tile [16][lda], B from LDS [K][ldw].
__device__ __forceinline__ v8f wmma_mm(const float* A, int lda,
                                       const float* Wt, int ldw,
                                       int kChunks, int nc, int hf)
{
  v8f acc = {0.f, 0.f, 0.f, 0.f, 0.f, 0.f, 0.f, 0.f};
  for (int c = 0; c < kChunks; ++c) {
    const int k0 = 4 * c + 2 * hf;
    v2f a, b;
    a.x = A[nc * lda + k0];
    a.y = A[nc * lda + k0 + 1];
    b.x = Wt[k0 * ldw + nc];
    b.y = Wt[(k0 + 1) * ldw + nc];
    acc = __builtin_amdgcn_wmma_f32_16x16x4_f32(false, a, false, b,
                                                (short)0, acc, false, false);
  }
  return acc;
}

__global__ __launch_bounds__(TPB)
void pinn_moe_fused(const float* __restrict__ tx,
                    const float* __restrict__ Wv,  const float* __restrict__ bv,
                    const float* __restrict__ Wo,  const float* __restrict__ bo,
                    const float* __restrict__ eW1, const float* __restrict__ eb1,
                    const float* __restrict__ eW2, const float* __restrict__ eb2,
                    const float* __restrict__ eW3, const float* __restrict__ eb3,
                    const float* __restrict__ eW4, const float* __restrict__ eb4,
                    const float* __restrict__ eW5, const float* __restrict__ eb5,
                    const float* __restrict__ Wp1, const float* __restrict__ bp1,
                    const float* __restrict__ Wp2, const float* __restrict__ bp2,
                    const float* __restrict__ Wp3, const float* __restrict__ bp3,
                    const float* __restrict__ pe2p,
                    float* __restrict__ dout, long nb)
{
  __shared__ float smem[WTOT + WAVES * WAVE_SCR];
  const int tid = threadIdx.x;

  // ---- stage weights in LDS (zero-padded); block-wide barriers here only ----
  for (int i = tid; i < WTOT; i += TPB) smem[i] = 0.f;
  __syncthreads();

  for (int i = tid; i < 3*12*4; i += TPB){int e=i/48, r=(i/4)%12, c=i%4;  smem[EW1+e*192+r*16+c]=eW1[i];}
  for (int i = tid; i < 3*4;    i += TPB){int e=i/4,  c=i%4;              smem[EB1+e*16+c]=eb1[i];}
  for (int i = tid; i < 3*4*8;  i += TPB){int e=i/32, r=(i/8)%4,  c=i%8;  smem[EW2+e*64 +r*16+c]=eW2[i];}
  for (int i = tid; i < 3*8;    i += TPB){int e=i/8,  c=i%8;              smem[EB2+e*16+c]=eb2[i];}
  for (int i = tid; i < 3*8*16; i += TPB){int e=i/128,r=(i/16)%8, c=i%16; smem[EW3+e*128+r*16+c]=eW3[i];}
  for (int i = tid; i < 3*16;   i += TPB){int e=i/16, c=i%16;             smem[EB3+e*16+c]=eb3[i];}
  for (int i = tid; i < 3*16*8; i += TPB){int e=i/128,r=(i/8)%16, c=i%8;  smem[EW4+e*256+r*16+c]=eW4[i];}
  for (int i = tid; i < 3*8;    i += TPB){int e=i/8,  c=i%8;              smem[EB4+e*16+c]=eb4[i];}
  for (int i = tid; i < 3*8*2;  i += TPB){int e=i/16, r=(i/2)%8,  c=i%2;  smem[EW5+e*128+r*16+c]=eW5[i];}
  for (int i = tid; i < 3*2;    i += TPB){int e=i/2,  c=i%2;              smem[EB5+e*16+c]=eb5[i];}
  for (int i = tid; i < 6*32;   i += TPB){int r=i/32, c=i%32;             smem[WP1+r*32+c]=Wp1[i];}
  for (int i = tid; i < 32;     i += TPB) smem[BP1+i]=bp1[i];
  for (int i = tid; i < 32*16;  i += TPB){int r=i/16, c=i%16;             smem[WP2+r*16+c]=Wp2[i];}
  for (int i = tid; i < 16;     i += TPB) smem[BP2+i]=bp2[i];
  for (int i = tid; i < 16;     i += TPB) smem[WP3+i*16]=Wp3[i];
  if (tid == 0) smem[BP3] = bp3[0];
  // fused attention: Wc = Wv@Wo (10x11), bc = bv@Wo + bo
  for (int i = tid; i < 110; i += TPB) {
    int r = i / 11, c = i % 11; float s = 0.f;
    for (int k = 0; k < 96; ++k) s += Wv[r*96 + k] * Wo[k*11 + c];
    smem[WC + r*16 + c] = s;
  }
  for (int i = tid; i < 11; i += TPB) {
    float s = bo[i];
    for (int k = 0; k < 96; ++k) s += bv[k] * Wo[k*11 + i];
    smem[BCV + i] = s;
  }
  __syncthreads();   // weights visible to all waves; read-only from here on

  const int wid  = tid >> 5;
  const int lane = tid & 31;
  const int nc   = lane & 15;  // A row / B col / D col
  const int hf   = lane >> 4;  // half-wave
  float* Wl = smem;
  float* TA = smem + WTOT + wid * WAVE_SCR; // 16x17 ping
  float* TB = TA + 272;                     // 16x17 pong
  float* TW = TA + 544;                     // z tile (16x17), later 16x33 SOH tile
  float* SC = TA + 1072;                    // per-sample scalars [8][16]
  const long base = (long)blockIdx.x * ROWS_PB + wid * 16;
  __builtin_prefetch(tx + base * 11, 0, 3);

  // ---- load xf tile (16 x 10, zero-padded to 16) ----
  for (int i = lane; i < 256; i += 32) {
    int m = i >> 4, k = i & 15;
    TA[m*17 + k] = (k < 10) ? tx[(base + m) * 11 + 1 + k] : 0.f;
  }
  wave_sync();

  // ---- xd = xf @ Wc + bc ; build z = [t, xd, 0...] in TW ----
  {
    v8f acc = wmma_mm(TA, 17, Wl + WC, 16, 3, nc, hf);
    float bb = Wl[BCV + nc];
#pragma unroll
    for (int v = 0; v < 8; ++v) {
      int m = 8 * hf + v;
      if (nc < 11) TW[m*17 + nc + 1] = acc[v] + bb;
    }
  }
  for (int i = lane; i < 16; i += 32) {
    TW[i*17 + 0]  = tx[(base + i) * 11];
    TW[i*17 + 12] = 0.f; TW[i*17 + 13] = 0.f;
    TW[i*17 + 14] = 0.f; TW[i*17 + 15] = 0.f;
  }
  wave_sync();

  const float pe2 = pe2p[0];

  // ---- experts: primal (capture relu masks in regs) + tangent wrt t ----
  for (int e = 0; e < 3; ++e) {
    const float* w1 = Wl+EW1+e*192; const float* b1 = Wl+EB1+e*16;
    const float* w2 = Wl+EW2+e*64;  const float* b2 = Wl+EB2+e*16;
    const float* w3 = Wl+EW3+e*128; const float* b3 = Wl+EB3+e*16;
    const float* w4 = Wl+EW4+e*256; const float* b4 = Wl+EB4+e*16;
    const float* w5 = Wl+EW5+e*128; const float* b5 = Wl+EB5+e*16;
    unsigned M1 = 0, M2 = 0, M3 = 0, M4 = 0;

    { v8f acc = wmma_mm(TW,17,w1,16,3,nc,hf); float bb=b1[nc];   // L1: z -> 4
#pragma unroll
      for (int v=0; v<8; ++v){ float p=acc[v]+bb; int m=8*hf+v;
        if (p>0.f) M1 |= 1u<<v; TA[m*17+nc] = p>0.f ? p : 0.f; } }
    wave_sync();
    { v8f acc = wmma_mm(TA,17,w2,16,1,nc,hf); float bb=b2[nc];   // L2: 4 -> 8
#pragma unroll
      for (int v=0; v<8; ++v){ float p=acc[v]+bb; int m=8*hf+v;
        if (p>0.f) M2 |= 1u<<v; TB[m*17+nc] = p>0.f ? p : 0.f; } }
    wave_sync();
    { v8f acc = wmma_mm(TB,17,w3,16,2,nc,hf); float bb=b3[nc];   // L3: 8 -> 16
#pragma unroll
      for (int v=0; v<8; ++v){ float p=acc[v]+bb; int m=8*hf+v;
        if (p>0.f) M3 |= 1u<<v; TA[m*17+nc] = p>0.f ? p : 0.f; } }
    wave_sync();
    { v8f acc = wmma_mm(TA,17,w4,16,4,nc,hf); float bb=b4[nc];   // L4: 16 -> 8
#pragma unroll
      for (int v=0; v<8; ++v){ float p=acc[v]+bb; int m=8*hf+v;
        if (p>0.f) M4 |= 1u<<v; TB[m*17+nc] = p>0.f ? p : 0.f; } }
    wave_sync();
    { v8f acc = wmma_mm(TB,17,w5,16,2,nc,hf); float bb=b5[nc];   // L5: 8 -> 2 (no relu)
#pragma unroll
      for (int v=0; v<8; ++v){ int m=8*hf+v;
        if (nc < 2) SC[(e*2 + nc)*16 + m] = acc[v] + bb; } }
    wave_sync();

    if (e < 2) {  // tangent d/dt: seed = M1 .* W1[0,:] (dz = e0), no biases
      float w1r0 = w1[nc];
#pragma unroll
      for (int v=0; v<8; ++v){ int m=8*hf+v; TA[m*17+nc] = ((M1>>v)&1u) ? w1r0 : 0.f; }
      wave_sync();
      { v8f acc = wmma_mm(TA,17,w2,16,1,nc,hf);
#pragma unroll
        for (int v=0; v<8; ++v){ int m=8*hf+v; TB[m*17+nc] = ((M2>>v)&1u) ? acc[v] : 0.f; } }
      wave_sync();
      { v8f acc = wmma_mm(TB,17,w3,16,2,nc,hf);
#pragma unroll
        for (int v=0; v<8; ++v){ int m=8*hf+v; TA[m*17+nc] = ((M3>>v)&1u) ? acc[v] : 0.f; } }
      wave_sync();
      { v8f acc = wmma_mm(TA,17,w4,16,4,nc,hf);
#pragma unroll
        for (int v=0; v<8; ++v){ int m=8*hf+v; TB[m*17+nc] = ((M4>>v)&1u) ? acc[v] : 0.f; } }
      wave_sync();
      { v8f acc = wmma_mm(TB,17,w5,16,2,nc,hf);
#pragma unroll
        for (int v=0; v<8; ++v){ int m=8*hf+v;
          if (e == 0 && nc == 0) SC[96  + m] = acc[v];   // T_t
          if (e == 1 && nc == 1) SC[112 + m] = acc[v]; } } // c_t
      wave_sync();
    }
  }

  // ---- per-sample scalars, losses, direct outputs ----
  // ReLU MLP => exact Hessian (lap) is 0: phi_lap = c_lap = 0; T_xx = sigma_x = 0.
  if (lane < 16) {
    int m = lane; long s = base + m;
    float T  = SC[0  + m], Q  = SC[16 + m];
    float ph = SC[32 + m], cc = SC[48 + m];
    float sg = SC[64 + m], ff = SC[80 + m];
    float Tt = SC[96 + m], ct = SC[112 + m];
    float rh = Tt + Q;          // res_heat
    float re = ct - pe2;        // res_elec (c_lap == 0)
    float loss = re*re + rh*rh + ff*ff;
    dout[nb*1 + s] = loss;
    dout[nb*2 + s] = ph;
    dout[nb*3 + s] = cc;
    dout[nb*4 + s] = T;
    dout[nb*5 + s] = Q;
    dout[nb*6 + s] = sg;
    dout[nb*7 + s] = ff;
  }

  // ---- SOH head: F_in = [phi, c, T, Qsrc, sigma, f] -> 32 -> 16 -> 1 ----
  for (int i = lane; i < 256; i += 32) {
    int m = i >> 4, k = i & 15; float v = 0.f;
    switch (k) {
      case 0: v = SC[32 + m]; break;  case 1: v = SC[48 + m]; break;
      case 2: v = SC[0  + m]; break;  case 3: v = SC[16 + m]; break;
      case 4: v = SC[64 + m]; break;  case 5: v = SC[80 + m]; break;
      default: v = 0.f; break;
    }
    TA[m*17 + k] = v;
  }
  wave_sync();
  { // P1: 6->32 (two N-halves), relu, into wide tile TW[16][33]
    v8f a0 = wmma_mm(TA, 17, Wl + WP1,      32, 2, nc, hf);
    v8f a1 = wmma_mm(TA, 17, Wl + WP1 + 16, 32, 2, nc, hf);
    float bb0 = Wl[BP1 + nc], bb1 = Wl[BP1 + 16 + nc];
#pragma unroll
    for (int v = 0; v < 8; ++v) {
      int m = 8*hf + v;
      float p0 = a0[v] + bb0, p1 = a1[v] + bb1;
      TW[m*33 + nc]      = p0 > 0.f ? p0 : 0.f;
      TW[m*33 + 16 + nc] = p1 > 0.f ? p1 : 0.f;
    }
  }
  wave_sync();
  { // P2: 32->16, relu
    v8f acc = wmma_mm(TW, 33, Wl + WP2, 16, 8, nc, hf);
    float bb = Wl[BP2 + nc];
#pragma unroll
    for (int v = 0; v < 8; ++v) { int m = 8*hf + v; float p = acc[v] + bb;
      TB[m*17 + nc] = p > 0.f ? p : 0.f; }
  }
  wave_sync();
  { // P3: 16->1, write soh
    v8f acc = wmma_mm(TB, 17, Wl + WP3, 16, 4, nc, hf);
    float bb = Wl[BP3];
#pragma unroll
    for (int v = 0; v < 8; ++v) { int m = 8*hf + v;
      if (nc == 0) dout[base + m] = acc[v] + bb; }
  }
}

extern "C" void kernel_launch(void* const* d_in, const int* in_sizes, int n_in,
                              void* d_out, int out_size, void* d_ws, size_t ws_size,
                              hipStream_t stream) {
  (void)n_in; (void)out_size; (void)d_ws; (void)ws_size;
  const float* tx  = (const float*)d_in[0];
  const float* Wv  = (const float*)d_in[5];
  const float* bv  = (const float*)d_in[6];
  const float* Wo  = (const float*)d_in[7];
  const float* bo  = (const float*)d_in[8];
  const float* eW1 = (const float*)d_in[9];
  const float* eb1 = (const float*)d_in[10];
  const float* eW2 = (const float*)d_in[11];
  const float* eb2 = (const float*)d_in[12];
  const float* eW3 = (const float*)d_in[13];
  const float* eb3 = (const float*)d_in[14];
  const float* eW4 = (const float*)d_in[15];
  const float* eb4 = (const float*)d_in[16];
  const float* eW5 = (const float*)d_in[17];
  const float* eb5 = (const float*)d_in[18];
  const float* Wp1 = (const float*)d_in[19];
  const float* bp1 = (const float*)d_in[20];
  const float* Wp2 = (const float*)d_in[21];
  const float* bp2 = (const float*)d_in[22];
  const float* Wp3 = (const float*)d_in[23];
  const float* bp3 = (const float*)d_in[24];
  const float* pe2 = (const float*)d_in[27];
  const long nb = in_sizes[0] / 11;           // B (131072)
  dim3 grid((unsigned)(nb / ROWS_PB)), block(TPB);
  hipLaunchKernelGGL(pinn_moe_fused, grid, block, 0, stream,
                     tx, Wv, bv, Wo, bo,
                     eW1, eb1, eW2, eb2, eW3, eb3, eW4, eb4, eW5, eb5,
                     Wp1, bp1, Wp2, bp2, Wp3, bp3, pe2,
                     (float*)d_out, nb);
}